// CustomLoss_88519275970662
// MI455X (gfx1250) — compile-verified
//
#include <hip/hip_runtime.h>

typedef float f4  __attribute__((ext_vector_type(4)));
typedef float v2f __attribute__((ext_vector_type(2)));
typedef float v8f __attribute__((ext_vector_type(8)));

// logaddexp(0,x) - x*t  computed stably: max(x,0) + log1p(exp(-|x|)) - x*t
__device__ __forceinline__ float seg_term(float x, float t) {
    return fmaxf(x, 0.f) + log1pf(__expf(-fabsf(x))) - x * t;
}

// One block per (b,d) slice of 256x256. 256 threads = 8 waves (wave32).
__global__ __launch_bounds__(256) void slice_kernel(
        const float* __restrict__ xo_all,   // model_output
        const float* __restrict__ tm_all,   // target_masks
        const float* __restrict__ tb_all,   // target_bboxes (B*D*4)
        float* __restrict__ segPart,        // [256]
        float* __restrict__ bboxPart) {     // [256]
    __shared__ unsigned rowmask[8];   // 256 row-occupancy bits
    __shared__ unsigned colmask[8];   // 256 col-occupancy bits
    __shared__ float    waveSeg[8];   // per-wave seg partials

    const int t = threadIdx.x;
    const int s = blockIdx.x;                 // slice = b*64 + d
    if (t < 8) { rowmask[t] = 0u; colmask[t] = 0u; }
    __syncthreads();

    const int tx   = t & 63;                  // column group (4 columns)
    const int ty   = t >> 6;                  // row phase 0..3
    const int col0 = tx << 2;
    const float* xo = xo_all + (size_t)s * 65536u;
    const float* tm = tm_all + (size_t)s * 65536u;

    float    acc    = 0.f;
    unsigned colnib = 0u;                     // 4 column-any bits

    for (int w = 0; w < 8; ++w) {             // 8 chunks x 8 rows = 64 rows/thread
        unsigned rw = 0u;                     // row-any bits for rows in word w
        #pragma unroll
        for (int k2 = 0; k2 < 8; ++k2) {
            const int y   = ((w * 8 + k2) << 2) + ty;   // y>>5 == w  (since ty<4)
            const int idx = (y << 8) + col0;
            // streamed exactly once -> non-temporal b128 loads
            f4 xv = __builtin_nontemporal_load((const f4*)(xo + idx));
            f4 tv = __builtin_nontemporal_load((const f4*)(tm + idx));
            acc += seg_term(xv.x, tv.x);
            acc += seg_term(xv.y, tv.y);
            acc += seg_term(xv.z, tv.z);
            acc += seg_term(xv.w, tv.w);
            // sigmoid(x) > 0.5  <=>  x > 0
            unsigned nib = (xv.x > 0.f ? 1u : 0u) | (xv.y > 0.f ? 2u : 0u)
                         | (xv.z > 0.f ? 4u : 0u) | (xv.w > 0.f ? 8u : 0u);
            colnib |= nib;
            if (nib) rw |= 1u << (y & 31);
        }
        if (rw) atomicOr(&rowmask[w], rw);
    }
    if (colnib) atomicOr(&colmask[tx >> 3], colnib << ((tx & 7) << 2));

    // ---- per-wave reduction of `acc` on the matrix pipe ----
    // A (16x4) holds the 32 lane partials (K=0 for lanes 0-15, K=2 for 16-31),
    // B = ones(4x16): D[m][n] = acc[m] + acc[m+16]. Sum 8 C-VGPRs + xor-16.
    v2f a;  a.x  = acc; a.y  = 0.f;
    v2f b1; b1.x = 1.f; b1.y = 1.f;
    v8f c = {};
    c = __builtin_amdgcn_wmma_f32_16x16x4_f32(false, a, false, b1,
                                              (short)0, c, false, false);
    float sum8    = c[0] + c[1] + c[2] + c[3] + c[4] + c[5] + c[6] + c[7];
    float waveTot = sum8 + __shfl_xor(sum8, 16, 32);

    const int lane = t & 31, wid = t >> 5;
    if (lane == 0) waveSeg[wid] = waveTot;
    __syncthreads();

    if (t == 0) {
        float seg = 0.f;
        for (int i = 0; i < 8; ++i) seg += waveSeg[i];

        unsigned anyAll = 0u;
        for (int i = 0; i < 8; ++i) anyAll |= rowmask[i];

        float pb[4];
        if (!anyAll) {
            pb[0] = 0.f; pb[1] = 0.f; pb[2] = 256.f; pb[3] = 256.f;
        } else {
            int ymin = 0, ymax = 0, xmin = 0, xmax = 0;
            for (int i = 0; i < 8; ++i)
                if (rowmask[i]) { ymin = i * 32 + __builtin_ctz(rowmask[i]); break; }
            for (int i = 7; i >= 0; --i)
                if (rowmask[i]) { ymax = i * 32 + 31 - __builtin_clz(rowmask[i]); break; }
            for (int i = 0; i < 8; ++i)
                if (colmask[i]) { xmin = i * 32 + __builtin_ctz(colmask[i]); break; }
            for (int i = 7; i >= 0; --i)
                if (colmask[i]) { xmax = i * 32 + 31 - __builtin_clz(colmask[i]); break; }
            pb[0] = (float)xmin;
            pb[1] = (float)ymin;
            pb[2] = (float)(xmax - xmin);
            pb[3] = (float)(ymax - ymin);
        }
        const float* tb = tb_all + s * 4;
        float bl = 0.f;
        #pragma unroll
        for (int j = 0; j < 4; ++j) {
            float d  = pb[j] - tb[j];
            float ad = fabsf(d);
            bl += (ad < 1.f) ? 0.5f * d * d : ad - 0.5f;
        }
        segPart[s]  = seg;
        bboxPart[s] = bl;
    }
}

// Final reduction: one wave, fp64 accumulation of 256 slice partials.
__global__ __launch_bounds__(32) void finalize_kernel(
        const float* __restrict__ segPart,
        const float* __restrict__ bboxPart,
        float* __restrict__ out) {
    const int l = threadIdx.x;
    double s = 0.0, b = 0.0;
    for (int i = l; i < 256; i += 32) {
        s += (double)segPart[i];
        b += (double)bboxPart[i];
    }
    #pragma unroll
    for (int off = 16; off > 0; off >>= 1) {
        s += __shfl_xor(s, off, 32);
        b += __shfl_xor(b, off, 32);
    }
    if (l == 0) {
        out[0] = (float)(s * (1.0 / 16777216.0));  // mean over B*C*D*H*W
        out[1] = (float)(b * (1.0 / 1024.0));      // mean over B*D*4
    }
}

extern "C" void kernel_launch(void* const* d_in, const int* in_sizes, int n_in,
                              void* d_out, int out_size, void* d_ws, size_t ws_size,
                              hipStream_t stream) {
    const float* model_output  = (const float*)d_in[0];
    const float* target_masks  = (const float*)d_in[1];
    const float* target_bboxes = (const float*)d_in[2];
    float* out = (float*)d_out;

    float* segPart  = (float*)d_ws;        // 256 floats
    float* bboxPart = segPart + 256;       // 256 floats

    slice_kernel<<<256, 256, 0, stream>>>(model_output, target_masks,
                                          target_bboxes, segPart, bboxPart);
    finalize_kernel<<<1, 32, 0, stream>>>(segPart, bboxPart, out);
}